// TFLlamaAttention_42863773614625
// MI455X (gfx1250) — compile-verified
//
#include <hip/hip_runtime.h>
#include <hip/hip_bf16.h>

typedef __attribute__((ext_vector_type(2))) float v2f;
typedef __attribute__((ext_vector_type(8))) float v8f;
typedef unsigned int u32x4 __attribute__((ext_vector_type(4)));
typedef int i32x8 __attribute__((ext_vector_type(8)));
typedef int i32x4 __attribute__((ext_vector_type(4)));

#define SEQ    2048
#define HIDDEN 2048
#define HEADS  16
#define HD     128
#define F32MIN (-3.4028234663852886e38f)

#if defined(__has_builtin)
#  if __has_builtin(__builtin_amdgcn_tensor_load_to_lds)
#    define USE_TDM 1
#  else
#    define USE_TDM 0
#  endif
#else
#  define USE_TDM 0
#endif

static __device__ __forceinline__ v8f vzero8() {
  v8f z;
#pragma unroll
  for (int i = 0; i < 8; ++i) z[i] = 0.f;
  return z;
}

// D = A(16x4) * B(4x16) + C, f32, wave32
static __device__ __forceinline__ v8f wmma4(v2f a, v2f b, v8f c) {
  return __builtin_amdgcn_wmma_f32_16x16x4_f32(
      false, a, false, b, (short)0, c, false, false);
}

// low 32 bits of a flat pointer to __shared__ == LDS byte offset
static __device__ __forceinline__ uint32_t lds_off(const void* p) {
  return (uint32_t)(uintptr_t)p;
}

// async global -> LDS, 16B per lane, ASYNCcnt-tracked (GV addressing)
static __device__ __forceinline__ void async_ld_b128(const void* g,
                                                     uint32_t lds) {
  asm volatile("global_load_async_to_lds_b128 %0, %1, off"
               :
               : "v"(lds), "v"((uint64_t)(uintptr_t)g)
               : "memory");
}

static __device__ __forceinline__ void wait_async0() {
#if defined(__has_builtin) && __has_builtin(__builtin_amdgcn_s_wait_asynccnt)
  __builtin_amdgcn_s_wait_asynccnt(0);
#else
  asm volatile("s_wait_asynccnt 0x0" ::: "memory");
#endif
}

static __device__ __forceinline__ void wait_tensor0() {
#if defined(__has_builtin) && __has_builtin(__builtin_amdgcn_s_wait_tensorcnt)
  __builtin_amdgcn_s_wait_tensorcnt(0);
#else
  asm volatile("s_wait_tensorcnt 0x0" ::: "memory");
#endif
}

#if USE_TDM
// TDM: load a 2D f32 tile (128 wide x 32 high) from a row-major tensor
// (row stride = HIDDEN elements) into LDS, padding 4 dwords every 128 dwords
// so a 128-wide row lands with pitch 132 floats (bank-conflict free).
static __device__ __forceinline__ void tdm_load_tile_f32(uint32_t lds_byte,
                                                         const float* gsrc) {
  const uint64_t ga = (uint64_t)(uintptr_t)gsrc;
  u32x4 g0;
  g0[0] = 1u;                                   // count=1 (valid), user mode
  g0[1] = lds_byte;                             // lds_addr
  g0[2] = (uint32_t)ga;                         // global_addr[31:0]
  g0[3] = (uint32_t)((ga >> 32) & 0x1FFFFFFu)   // global_addr[56:32]
          | (2u << 30);                         // type = 2 ("image")
  i32x8 g1;
  // workgroup_mask=0, data_size=2 (4B), pad_enable=1,
  // pad_interval=6 (128 dwords), pad_amount=3 (4 dwords)
  g1[0] = (int)((2u << 16) | (1u << 20) | (6u << 22) | (3u << 25));
  const uint32_t tw = HIDDEN, th = SEQ;          // tensor dims (elements)
  const uint32_t tilew = 128u, tileh = 32u;      // tile dims
  g1[1] = (int)((tw & 0xFFFFu) << 16);           // tensor_dim0[15:0]
  g1[2] = (int)(((tw >> 16) & 0xFFFFu) | ((th & 0xFFFFu) << 16));
  g1[3] = (int)(((th >> 16) & 0xFFFFu) | ((tilew & 0xFFFFu) << 16));
  g1[4] = (int)(tileh & 0xFFFFu);                // tile_dim1, tile_dim2=0
  g1[5] = (int)HIDDEN;                           // tensor_dim0_stride[31:0]
  g1[6] = 0;
  g1[7] = 0;
  const i32x4 gz4 = {0, 0, 0, 0};
  const i32x8 gz8 = {0, 0, 0, 0, 0, 0, 0, 0};
  // 6-arg toolchain variant: (g0, g1, g2, g3, g4, cpol)
  __builtin_amdgcn_tensor_load_to_lds(g0, g1, gz4, gz4, gz8, 0);
}
#endif

// ---------------------------------------------------------------------------
// C[2048x2048] = A[2048x2048] * B[2048x2048], row-major f32.
// Block tile 128x128, K-tile 32, double-buffered LDS filled with
// GLOBAL_LOAD_ASYNC_TO_LDS_B128; 8 waves, wave tile 32x64 = 2x4 WMMA tiles.
// ---------------------------------------------------------------------------
__global__ __launch_bounds__(256) void gemm2048_wmma(
    const float* __restrict__ A, const float* __restrict__ B,
    float* __restrict__ C) {
  __shared__ float As[2][128][36];
  __shared__ float Bs[2][32][132];

  const int tid  = threadIdx.x;
  const int lane = tid & 31;
  const int wave = tid >> 5;
  const int hi   = lane >> 4;
  const int ln   = lane & 15;
  const int mwave = (wave >> 1) * 32;
  const int nwave = (wave & 1) * 64;
  const int mblk = blockIdx.y * 128;
  const int nblk = blockIdx.x * 128;

  v8f acc[2][4];
#pragma unroll
  for (int mt = 0; mt < 2; ++mt)
#pragma unroll
    for (int nt = 0; nt < 4; ++nt) acc[mt][nt] = vzero8();

  const int ar = tid >> 3;          // A: 8 threads/row of 32
  const int ac = (tid & 7) * 4;
  const int br = tid >> 5;          // B: 32 threads/row of 128
  const int bc = (tid & 31) * 4;

  auto prefetch = [&](int kt, int buf) {
#pragma unroll
    for (int p = 0; p < 4; ++p)
      async_ld_b128(&A[(size_t)(mblk + ar + p * 32) * HIDDEN + kt + ac],
                    lds_off(&As[buf][ar + p * 32][ac]));
#pragma unroll
    for (int p = 0; p < 4; ++p)
      async_ld_b128(&B[(size_t)(kt + br + p * 8) * HIDDEN + nblk + bc],
                    lds_off(&Bs[buf][br + p * 8][bc]));
  };

  prefetch(0, 0);
  int buf = 0;
  for (int kt = 0; kt < HIDDEN; kt += 32) {
    wait_async0();        // my async writes into As/Bs[buf] are done
    __syncthreads();      // everyone's done; prev reads of buf^1 done
    if (kt + 32 < HIDDEN) prefetch(kt + 32, buf ^ 1);  // overlap next DMA

#pragma unroll
    for (int ks = 0; ks < 8; ++ks) {
      const int kk = ks * 4 + hi * 2;
      v2f af[2], bf[4];
#pragma unroll
      for (int mt = 0; mt < 2; ++mt) {
        af[mt].x = As[buf][mwave + mt * 16 + ln][kk];
        af[mt].y = As[buf][mwave + mt * 16 + ln][kk + 1];
      }
#pragma unroll
      for (int nt = 0; nt < 4; ++nt) {
        bf[nt].x = Bs[buf][kk][nwave + nt * 16 + ln];
        bf[nt].y = Bs[buf][kk + 1][nwave + nt * 16 + ln];
      }
#pragma unroll
      for (int mt = 0; mt < 2; ++mt)
#pragma unroll
        for (int nt = 0; nt < 4; ++nt)
          acc[mt][nt] = wmma4(af[mt], bf[nt], acc[mt][nt]);
    }
    buf ^= 1;
  }

#pragma unroll
  for (int mt = 0; mt < 2; ++mt)
#pragma unroll
    for (int nt = 0; nt < 4; ++nt)
#pragma unroll
      for (int i = 0; i < 8; ++i) {
        const int row = mblk + mwave + mt * 16 + i + 8 * hi;
        const int col = nblk + nwave + nt * 16 + ln;
        C[(size_t)row * HIDDEN + col] = acc[mt][nt][i];
      }
}

// ---------------------------------------------------------------------------
// RoPE in place on Q and K.
// ---------------------------------------------------------------------------
__global__ void rope_kernel(float* __restrict__ Q, float* __restrict__ Km,
                            const int* __restrict__ pos_ids) {
  const int idx = blockIdx.x * blockDim.x + threadIdx.x;  // SEQ*HEADS*64
  const int s = idx >> 10;
  const int r = idx & 1023;
  const int h = r >> 6;
  const int j = r & 63;
  const float pos = (float)pos_ids[s];
  const float freq = __powf(10000.0f, -(float)j * (1.0f / 64.0f));
  const float ang = pos * freq;
  const float c = cosf(ang), sn = sinf(ang);
  const size_t base = (size_t)s * HIDDEN + h * HD + j;
  const float q0 = Q[base], q1 = Q[base + 64];
  Q[base]      = q0 * c - q1 * sn;
  Q[base + 64] = q1 * c + q0 * sn;
  const float k0 = Km[base], k1 = Km[base + 64];
  Km[base]      = k0 * c - k1 * sn;
  Km[base + 64] = k1 * c + k0 * sn;
}

// ---------------------------------------------------------------------------
// Flash attention, f32 WMMA. Grid = (SEQ/128, HEADS), 256 threads (8 waves).
// Each wave owns 16 query rows; 32-key blocks double-buffered in LDS and
// staged by the Tensor Data Mover (wave 0 issues, TENSORcnt-waited), with an
// async-LDS fallback when the TDM builtin is unavailable.
// ---------------------------------------------------------------------------
__global__ __launch_bounds__(256) void attn_fa_kernel(
    const float* __restrict__ Q, const float* __restrict__ K,
    const float* __restrict__ V, const float* __restrict__ mask,
    float* __restrict__ O) {
  __shared__ float Ks[2][32][132];
  __shared__ float Vs[2][32][132];
  __shared__ float Ps[8][16][36];

  const int tid  = threadIdx.x;
  const int lane = tid & 31;
  const int wave = tid >> 5;
  const int hi   = lane >> 4;
  const int ln   = lane & 15;
  const int h    = blockIdx.y;
  const int qbase = blockIdx.x * 128 + wave * 16;
  const float scale = 0.0883883476483184f;  // 1/sqrt(128)

  // Q fragments (A-layout: M=ln, K split across lane halves)
  v2f qf[32];
  {
    const float* qrow = Q + (size_t)(qbase + ln) * HIDDEN + h * HD;
#pragma unroll
    for (int ks = 0; ks < 32; ++ks) {
      const int kk = ks * 4 + hi * 2;
      qf[ks].x = qrow[kk];
      qf[ks].y = qrow[kk + 1];
    }
  }

  v8f o[8];
#pragma unroll
  for (int nt = 0; nt < 8; ++nt) o[nt] = vzero8();
  float m[8], l[8];
#pragma unroll
  for (int i = 0; i < 8; ++i) { m[i] = F32MIN; l[i] = 0.f; }

  const int r = tid >> 3;
  const int cbase = (tid & 7) * 16;

  auto stage = [&](int kb, int buf) {
#if USE_TDM
    if (wave == 0) {
      tdm_load_tile_f32(lds_off(&Ks[buf][0][0]),
                        K + (size_t)(kb * 32) * HIDDEN + h * HD);
      tdm_load_tile_f32(lds_off(&Vs[buf][0][0]),
                        V + (size_t)(kb * 32) * HIDDEN + h * HD);
    }
#else
    const float* kr = K + (size_t)(kb * 32 + r) * HIDDEN + h * HD + cbase;
    const float* vr = V + (size_t)(kb * 32 + r) * HIDDEN + h * HD + cbase;
#pragma unroll
    for (int u = 0; u < 4; ++u) {
      async_ld_b128(kr + u * 4, lds_off(&Ks[buf][r][cbase + u * 4]));
      async_ld_b128(vr + u * 4, lds_off(&Vs[buf][r][cbase + u * 4]));
    }
#endif
  };
  auto stage_wait = [&]() {
#if USE_TDM
    if (wave == 0) wait_tensor0();
#else
    wait_async0();
#endif
    __syncthreads();
  };

  stage(0, 0);
  int buf = 0;
  for (int kb = 0; kb < SEQ / 32; ++kb) {
    stage_wait();
    if (kb + 1 < SEQ / 32) stage(kb + 1, buf ^ 1);  // DMA next block now

    // S = Q(16x128) * K^T(128x32)
    v8f s0 = vzero8(), s1 = vzero8();
#pragma unroll
    for (int ks = 0; ks < 32; ++ks) {
      const int kk = ks * 4 + hi * 2;
      v2f b0, b1;
      b0.x = Ks[buf][ln][kk];      b0.y = Ks[buf][ln][kk + 1];
      b1.x = Ks[buf][16 + ln][kk]; b1.y = Ks[buf][16 + ln][kk + 1];
      s0 = wmma4(qf[ks], b0, s0);
      s1 = wmma4(qf[ks], b1, s1);
    }

    // mask + online softmax (row i+8*hi lives in this lane's half-wave)
    float p0[8], p1[8], rmax[8], rsum[8], corr[8];
#pragma unroll
    for (int i = 0; i < 8; ++i) {
      const size_t moff = (size_t)(qbase + i + 8 * hi) * SEQ + (size_t)kb * 32;
      const float sv0 = fmaxf(s0[i] * scale + mask[moff + ln], F32MIN);
      const float sv1 = fmaxf(s1[i] * scale + mask[moff + 16 + ln], F32MIN);
      p0[i] = sv0;
      p1[i] = sv1;
      rmax[i] = fmaxf(m[i], fmaxf(sv0, sv1));
    }
#pragma unroll
    for (int off = 8; off >= 1; off >>= 1)
#pragma unroll
      for (int i = 0; i < 8; ++i)
        rmax[i] = fmaxf(rmax[i], __shfl_xor(rmax[i], off, 32));
#pragma unroll
    for (int i = 0; i < 8; ++i) {
      corr[i] = __expf(m[i] - rmax[i]);
      m[i] = rmax[i];
      const float e0 = __expf(p0[i] - m[i]);
      const float e1 = __expf(p1[i] - m[i]);
      p0[i] = e0;
      p1[i] = e1;
      rsum[i] = e0 + e1;
    }
#pragma unroll
    for (int off = 8; off >= 1; off >>= 1)
#pragma unroll
      for (int i = 0; i < 8; ++i) rsum[i] += __shfl_xor(rsum[i], off, 32);
#pragma unroll
    for (int i = 0; i < 8; ++i) l[i] = l[i] * corr[i] + rsum[i];
#pragma unroll
    for (int nt = 0; nt < 8; ++nt)
#pragma unroll
      for (int i = 0; i < 8; ++i) o[nt][i] *= corr[i];

    // P (C-layout) -> per-wave LDS scratch -> A-fragments
#pragma unroll
    for (int i = 0; i < 8; ++i) {
      Ps[wave][i + 8 * hi][ln] = p0[i];
      Ps[wave][i + 8 * hi][16 + ln] = p1[i];
    }
    // O += P(16x32) * V(32x128)
#pragma unroll
    for (int ks = 0; ks < 8; ++ks) {
      const int kk = ks * 4 + hi * 2;
      v2f a;
      a.x = Ps[wave][ln][kk];
      a.y = Ps[wave][ln][kk + 1];
#pragma unroll
      for (int nt = 0; nt < 8; ++nt) {
        v2f b;
        b.x = Vs[buf][kk][nt * 16 + ln];
        b.y = Vs[buf][kk + 1][nt * 16 + ln];
        o[nt] = wmma4(a, b, o[nt]);
      }
    }
    buf ^= 1;
  }

#pragma unroll
  for (int nt = 0; nt < 8; ++nt)
#pragma unroll
    for (int i = 0; i < 8; ++i) {
      const int row = qbase + i + 8 * hi;
      O[(size_t)row * HIDDEN + h * HD + nt * 16 + ln] = o[nt][i] / l[i];
    }
}

// ---------------------------------------------------------------------------
extern "C" void kernel_launch(void* const* d_in, const int* in_sizes, int n_in,
                              void* d_out, int out_size, void* d_ws,
                              size_t ws_size, hipStream_t stream) {
  const float* X     = (const float*)d_in[0];  // hidden_states [1,2048,2048]
  const float* maskp = (const float*)d_in[1];  // [1,1,2048,2048]
  const int*   pos   = (const int*)d_in[2];    // [1,2048]
  const float* Wq    = (const float*)d_in[3];
  const float* Wk    = (const float*)d_in[4];
  const float* Wv    = (const float*)d_in[5];
  const float* Wo    = (const float*)d_in[6];
  float* out = (float*)d_out;

  float* ws = (float*)d_ws;
  const size_t MAT = (size_t)SEQ * HIDDEN;  // 16 MB per buffer
  float* Qb = ws;
  float* Kb = ws + MAT;
  float* Vb = ws + 2 * MAT;
  float* Ab = ws + 3 * MAT;

  const dim3 gGemm(HIDDEN / 128, SEQ / 128);
  const dim3 blk(256);

  gemm2048_wmma<<<gGemm, blk, 0, stream>>>(X, Wq, Qb);
  gemm2048_wmma<<<gGemm, blk, 0, stream>>>(X, Wk, Kb);
  gemm2048_wmma<<<gGemm, blk, 0, stream>>>(X, Wv, Vb);

  rope_kernel<<<(SEQ * HEADS * 64) / 256, blk, 0, stream>>>(Qb, Kb, pos);

  attn_fa_kernel<<<dim3(SEQ / 128, HEADS), blk, 0, stream>>>(Qb, Kb, Vb,
                                                             maskp, Ab);

  gemm2048_wmma<<<gGemm, blk, 0, stream>>>(Ab, Wo, out);
}